// GatedDeltaRuleModule_26792005992796
// MI455X (gfx1250) — compile-verified
//
#include <hip/hip_runtime.h>
#include <hip/hip_bf16.h>
#include <math.h>

// CDNA5 / gfx1250: banded gated attention (window 4) for B=2,S=8192,H=16,D=128.
// Bandwidth-bound (~540MB @ 23.3TB/s => ~23us floor). WMMA f32 16x16x4 does the
// QK^T scores and the (gated W) x V output matmuls; LDS transposes W from the
// C/D layout into the A-operand layout. Tile body is specialized on FIRST
// (tile 0) so the hot loops are branch-free and loads can be batched ahead of
// the WMMAs (staggered s_wait_loadcnt instead of drain-to-zero per step).

typedef __attribute__((ext_vector_type(2))) float v2f;
typedef __attribute__((ext_vector_type(8))) float v8f;

constexpr int Bc = 2, Sc = 8192, Hc = 16, Dc = 128, Wc = 4;
constexpr int TILE = 16;
constexpr int NT = Sc / TILE;              // 512 tiles along S
constexpr int WAVES_PER_BLOCK = 4;
constexpr int LDS_ROW = 18;                // padded row stride (floats): b64-aligned, conflict-free
constexpr float SCALE = 0.08838834764831845f;  // 1/sqrt(128)

__device__ __forceinline__ v8f wmma_f32(v2f a, v2f b, v8f c) {
    return __builtin_amdgcn_wmma_f32_16x16x4_f32(false, a, false, b,
                                                 (short)0, c, false, false);
}

template<bool FIRST>
__device__ __forceinline__ void process_tile(
    const float* __restrict__ q, const float* __restrict__ k,
    const float* __restrict__ v, const float* __restrict__ ag,
    const float* __restrict__ bg, float* __restrict__ out,
    float* wcur_lds, float* wprv_lds,
    int b, int h, int t0, int row, int khalf)
{
    const size_t rstride = (size_t)Hc * Dc;  // float stride between consecutive s positions

    // ------------------------------------------------------------------
    // Phase 1: scores S_cur = Q K_cur^T (and S_prv = Q K_prv^T if !FIRST).
    // A layout (16x4 f32): lane(M=row) holds K = off + {0,1}  -> contiguous float2
    // B layout (4x16 f32): lane(N=row) holds Kmat[j=row][off + {0,1}] float2
    // Loads are batched 16-24 deep per chunk before the WMMA burst.
    // ------------------------------------------------------------------
    const float* qrow = q + ((size_t)(b * Sc + t0 + row) * Hc + h) * Dc;
    const float* kcur = k + ((size_t)(b * Sc + t0 + row) * Hc + h) * Dc;
    const float* kprv = kcur - TILE * rstride;

    v8f s_cur = {0.f,0.f,0.f,0.f,0.f,0.f,0.f,0.f};
    v8f s_prv = {0.f,0.f,0.f,0.f,0.f,0.f,0.f,0.f};

    for (int kk0 = 0; kk0 < Dc; kk0 += 32) {   // 4 chunks x 8 K-steps
        v2f aq[8], bc[8], bp[8];
        #pragma unroll
        for (int u = 0; u < 8; ++u) {
            const int off = kk0 + 4 * u + 2 * khalf;
            aq[u] = *(const v2f*)(qrow + off);
            bc[u] = *(const v2f*)(kcur + off);
            if (!FIRST) bp[u] = *(const v2f*)(kprv + off);
        }
        #pragma unroll
        for (int u = 0; u < 8; ++u) {
            s_cur = wmma_f32(aq[u], bc[u], s_cur);
            if (!FIRST) s_prv = wmma_f32(aq[u], bp[u], s_prv);
        }
    }

    // ------------------------------------------------------------------
    // Phase 2: band mask 0 <= i-j <= W, scale, gate sigmoid(a_i*b_j);
    // stash W row-major in LDS for the A-operand reload.
    // C/D layout: lane holds N=row; VGPR r holds M = r + 8*khalf.
    // ------------------------------------------------------------------
    const float bj_cur = bg[(size_t)(b * Sc + t0 + row) * Hc + h];
    const float bj_prv = FIRST ? 0.0f
                               : bg[(size_t)(b * Sc + (t0 - TILE) + row) * Hc + h];

    #pragma unroll
    for (int r = 0; r < 8; ++r) {
        const int M = r + 8 * khalf;
        const float ai = ag[(size_t)(b * Sc + t0 + M) * Hc + h];
        {   // current tile: j = t0 + row  ->  delta = M - row
            const int d = M - row;
            const bool ok = (d >= 0) && (d <= Wc);
            const float g = 1.0f / (1.0f + __expf(-ai * bj_cur));
            float wv = ok ? (s_cur[r] * SCALE * g) : 0.0f;
            if (FIRST && M == 0) wv = 0.0f;   // position 0: out = v_0 (substituted at store)
            wcur_lds[M * LDS_ROW + row] = wv;
        }
        if (!FIRST) {  // previous tile: j = t0 - 16 + row -> delta = M - row + 16
            const int d = M - row + TILE;
            const bool ok = (d >= 0) && (d <= Wc);
            const float g = 1.0f / (1.0f + __expf(-ai * bj_prv));
            wprv_lds[M * LDS_ROW + row] = ok ? (s_prv[r] * SCALE * g) : 0.0f;
        }
    }

    // Cross-lane LDS dependency within the wave: drain DS before reload.
    asm volatile("s_wait_dscnt 0x0" ::: "memory");

    // ------------------------------------------------------------------
    // Phase 3: O = W_prv * V_prv + W_cur * V_cur   (16x16 by 16x128).
    // W fragments (A layout) are n-invariant: load from LDS once.
    // ------------------------------------------------------------------
    v2f wa_c[4], wa_p[4];
    #pragma unroll
    for (int ks = 0; ks < 4; ++ks) {
        const int kidx = 4 * ks + 2 * khalf;
        wa_c[ks] = *(const v2f*)(wcur_lds + row * LDS_ROW + kidx);
        if (!FIRST) wa_p[ks] = *(const v2f*)(wprv_lds + row * LDS_ROW + kidx);
    }

    const float* vcur = v + ((size_t)(b * Sc + t0) * Hc + h) * Dc;
    const float* vprv = vcur - TILE * rstride;
    float* orow = out + ((size_t)(b * Sc + t0) * Hc + h) * Dc;

    for (int n = 0; n < 8; ++n) {
        const int dcol = n * 16 + row;

        // Batch all V fragments for this column tile before the WMMA burst.
        v2f vb_c[4], vb_p[4];
        #pragma unroll
        for (int ks = 0; ks < 4; ++ks) {
            const int kidx = 4 * ks + 2 * khalf;
            vb_c[ks].x = vcur[(size_t)(kidx + 0) * rstride + dcol];
            vb_c[ks].y = vcur[(size_t)(kidx + 1) * rstride + dcol];
            if (!FIRST) {
                vb_p[ks].x = vprv[(size_t)(kidx + 0) * rstride + dcol];
                vb_p[ks].y = vprv[(size_t)(kidx + 1) * rstride + dcol];
            }
        }

        v8f o = {0.f,0.f,0.f,0.f,0.f,0.f,0.f,0.f};
        #pragma unroll
        for (int ks = 0; ks < 4; ++ks) {
            if (!FIRST) o = wmma_f32(wa_p[ks], vb_p[ks], o);
            o = wmma_f32(wa_c[ks], vb_c[ks], o);
        }

        // Store: lanes 0-15 write row t0+r, lanes 16-31 row t0+r+8.
        #pragma unroll
        for (int r = 0; r < 8; ++r) {
            const int M = r + 8 * khalf;
            float val = o[r];
            if (FIRST && M == 0) val = vcur[dcol];   // out[:,0] = v[:,0]
            orow[(size_t)M * rstride + dcol] = val;
        }
    }
}

__global__ __launch_bounds__(WAVES_PER_BLOCK * 32)
void gdr_banded_wmma(const float* __restrict__ q,
                     const float* __restrict__ k,
                     const float* __restrict__ v,
                     const float* __restrict__ ag,
                     const float* __restrict__ bg,
                     float* __restrict__ out)
{
    // Per-wave private LDS: gated weight tiles W_cur, W_prev (16 x 18 floats each)
    __shared__ float lds[WAVES_PER_BLOCK][2][TILE * LDS_ROW];

    const int lane  = threadIdx.x & 31;
    const int wave  = threadIdx.x >> 5;
    const int gw    = blockIdx.x * WAVES_PER_BLOCK + wave;   // global wave id
    const int tile  = gw % NT;
    const int h     = (gw / NT) % Hc;
    const int b     = gw / (NT * Hc);

    const int t0    = tile * TILE;
    const int row   = lane & 15;    // A-operand: M row.  B-operand / C,D: N column.
    const int khalf = lane >> 4;    // which half of the K-pairs this lane holds

    float* wcur_lds = lds[wave][0];
    float* wprv_lds = lds[wave][1];

    if (tile == 0) {
        process_tile<true >(q, k, v, ag, bg, out, wcur_lds, wprv_lds, b, h, t0, row, khalf);
    } else {
        process_tile<false>(q, k, v, ag, bg, out, wcur_lds, wprv_lds, b, h, t0, row, khalf);
    }
}

extern "C" void kernel_launch(void* const* d_in, const int* in_sizes, int n_in,
                              void* d_out, int out_size, void* d_ws, size_t ws_size,
                              hipStream_t stream) {
    const float* q  = (const float*)d_in[0];
    const float* k  = (const float*)d_in[1];
    const float* v  = (const float*)d_in[2];
    const float* ag = (const float*)d_in[3];
    const float* bg = (const float*)d_in[4];
    float* out = (float*)d_out;

    const int total_waves = Bc * Hc * NT;                    // 16384
    const int blocks = total_waves / WAVES_PER_BLOCK;        // 4096
    gdr_banded_wmma<<<blocks, WAVES_PER_BLOCK * 32, 0, stream>>>(q, k, v, ag, bg, out);
}